// LIFNeuron_32719060861203
// MI455X (gfx1250) — compile-verified
//
#include <hip/hip_runtime.h>

// LIF e-prop recurrent network, MI455X (gfx1250, wave32).
// Dims fixed by the reference.
constexpr int B = 64, T = 256, I = 256, H = 512, O = 128;

#define ALPHA 0.951229424500714f   // exp(-1/20)
#define KAPPA 0.951229424500714f   // exp(-1/20)
#define V_TH  1.0f
#define GAMMA 0.3f

typedef float v2f __attribute__((ext_vector_type(2)));
typedef float v4f __attribute__((ext_vector_type(4)));
typedef float v8f __attribute__((ext_vector_type(8)));

// ---------------------------------------------------------------------------
// Zero-init (state must be re-zeroed every call: harness poisons ws/out once
// and never restores between replays).
// ---------------------------------------------------------------------------
__global__ void lif_zero(float* __restrict__ p, int n /* multiple of 4 */) {
    int stride = gridDim.x * blockDim.x * 4;
    for (int i = (blockIdx.x * blockDim.x + threadIdx.x) * 4; i < n; i += stride) {
        *(v4f*)(p + i) = v4f{0.f, 0.f, 0.f, 0.f};
    }
}

// ---------------------------------------------------------------------------
// K1: i_t = x_t @ W_in^T + z_prev @ W_rec^T, fused with the neuron update.
// One wave32 owns one 16x16 tile of [B,H]; fp32 WMMA 16x16x4.
// z is double-buffered (every tile reads all of z_prev).
// ---------------------------------------------------------------------------
__global__ __launch_bounds__(256) void lif_gemm_neuron(
    const float* __restrict__ x_seq,   // [B,T,I]
    const float* __restrict__ W_in,    // [H,I]
    const float* __restrict__ W_rec,   // [H,H]
    const float* __restrict__ z_cur,   // [B,H]  (z_prev)
    float* __restrict__ z_nxt,         // [B,H]  (z_next)
    float* __restrict__ v,             // [B,H]
    float* __restrict__ psi,           // [B,H]
    int t) {
    const int wave = (blockIdx.x * blockDim.x + threadIdx.x) >> 5;
    const int lane = threadIdx.x & 31;
    const int tilesN = H / 16;                  // 32
    const int bm = (wave / tilesN) * 16;        // batch-tile row
    const int hn = (wave % tilesN) * 16;        // hidden-tile col
    const int half = lane >> 4;                 // 0: K=k..k+1, 1: K=k+2..k+3
    const int l16  = lane & 15;

    v8f acc = {};

    // ---- x_t @ W_in^T over K = I ----
    const float* xrow = x_seq + (size_t)(bm + l16) * T * I + (size_t)t * I + 2 * half;
    const float* wri  = W_in  + (size_t)(hn + l16) * I + 2 * half;
    for (int k = 0; k < I; k += 4) {
        v2f a = *(const v2f*)(xrow + k);
        v2f b = *(const v2f*)(wri + k);
        acc = __builtin_amdgcn_wmma_f32_16x16x4_f32(false, a, false, b,
                                                    (short)0, acc, false, false);
    }
    // ---- z_prev @ W_rec^T over K = H ----
    const float* zrow = z_cur + (size_t)(bm + l16) * H + 2 * half;
    const float* wrr  = W_rec + (size_t)(hn + l16) * H + 2 * half;
    for (int k = 0; k < H; k += 4) {
        v2f a = *(const v2f*)(zrow + k);
        v2f b = *(const v2f*)(wrr + k);
        acc = __builtin_amdgcn_wmma_f32_16x16x4_f32(false, a, false, b,
                                                    (short)0, acc, false, false);
    }

    // ---- neuron update on the 16x16 tile (lane holds rows r,r+8 by half) ----
#pragma unroll
    for (int r = 0; r < 8; ++r) {
        const int row = bm + r + 8 * half;
        const int col = hn + l16;
        const size_t idx = (size_t)row * H + col;
        const float zp = z_cur[idx];
        float vn = ALPHA * v[idx] + acc[r] - zp * V_TH;
        const float ps = (GAMMA / V_TH) * fmaxf(0.f, 1.f - fabsf(vn - V_TH) / V_TH);
        const float zn = (vn > V_TH) ? 1.f : 0.f;
        vn = vn - zn * V_TH;               // soft reset
        v[idx]     = vn;
        psi[idx]   = ps;
        z_nxt[idx] = zn;
    }
}

// ---------------------------------------------------------------------------
// K2: trace updates (must complete before K3 per the reference ordering).
// tr_in = a*tr_in + x_t ; tr_rec = a*tr_rec + z_prev
// ---------------------------------------------------------------------------
__global__ __launch_bounds__(256) void lif_trace(
    const float* __restrict__ x_seq, const float* __restrict__ z_cur,
    float* __restrict__ tr_in, float* __restrict__ tr_rec, int t) {
    const int nin  = B * I;        // 16384
    const int ntot = nin + B * H;  // 49152
    const int idx = blockIdx.x * blockDim.x + threadIdx.x;
    if (idx < nin) {
        const int b = idx / I, i = idx - b * I;
        tr_in[idx] = ALPHA * tr_in[idx] + x_seq[((size_t)b * T + t) * I + i];
    } else if (idx < ntot) {
        const int j = idx - nin;
        tr_rec[j] = ALPHA * tr_rec[j] + z_cur[j];
    }
}

// ---------------------------------------------------------------------------
// K3: eligibility traces — the bandwidth roofline (~200 MB r+w per step,
// ~100 MB working set -> L2 resident on MI455X's 192 MB L2).
// fe = KAPPA*fe + psi[b,h] * trace[b,:]; float4 vectorized grid-stride.
// ---------------------------------------------------------------------------
__global__ __launch_bounds__(256) void lif_fe(
    float* __restrict__ fe_in,            // [B,H,I]
    float* __restrict__ fe_rec,           // [B,H,H]
    const float* __restrict__ psi,        // [B,H]
    const float* __restrict__ tr_in,      // [B,I]
    const float* __restrict__ tr_rec) {   // [B,H]
    const int QIN  = B * H * (I / 4);     // 2,097,152 float4's
    const int QREC = B * H * (H / 4);     // 4,194,304 float4's
    const int stride = gridDim.x * blockDim.x;
    for (int q = blockIdx.x * blockDim.x + threadIdx.x; q < QIN + QREC; q += stride) {
        if (q < QIN) {
            const int bh = q / (I / 4);             // b*H + h
            const int i4 = q - bh * (I / 4);
            const int b  = bh >> 9;                 // / H
            const float p = psi[bh];
            const v4f tr = *(const v4f*)(tr_in + (size_t)b * I + i4 * 4);
            v4f f = *(v4f*)(fe_in + (size_t)q * 4);
            f = f * KAPPA + tr * p;
            *(v4f*)(fe_in + (size_t)q * 4) = f;
        } else {
            const int qr = q - QIN;
            const int bh = qr / (H / 4);
            const int h4 = qr - bh * (H / 4);
            const int b  = bh >> 9;
            const float p = psi[bh];
            const v4f tr = *(const v4f*)(tr_rec + (size_t)b * H + h4 * 4);
            v4f f = *(v4f*)(fe_rec + (size_t)qr * 4);
            f = f * KAPPA + tr * p;
            *(v4f*)(fe_rec + (size_t)qr * 4) = f;
        }
    }
}

// ---------------------------------------------------------------------------
// K4: ro = KAPPA*ro + z_next @ W_out^T ; store readout_stack[b,t,:].
// Wave per 16x16 tile of [B,O]; fp32 WMMA over K=H.
// ---------------------------------------------------------------------------
__global__ __launch_bounds__(256) void lif_readout(
    const float* __restrict__ z_nxt,   // [B,H]
    const float* __restrict__ W_out,   // [O,H]
    float* __restrict__ ro,            // [B,O]
    float* __restrict__ out,           // [B,T,O]
    int t) {
    const int wave = (blockIdx.x * blockDim.x + threadIdx.x) >> 5;
    const int lane = threadIdx.x & 31;
    const int tilesN = O / 16;                 // 8
    const int bm = (wave / tilesN) * 16;
    const int on = (wave % tilesN) * 16;
    const int half = lane >> 4;
    const int l16  = lane & 15;

    v8f acc = {};
    const float* zrow = z_nxt + (size_t)(bm + l16) * H + 2 * half;
    const float* wro  = W_out + (size_t)(on + l16) * H + 2 * half;
    for (int k = 0; k < H; k += 4) {
        v2f a = *(const v2f*)(zrow + k);
        v2f b = *(const v2f*)(wro + k);
        acc = __builtin_amdgcn_wmma_f32_16x16x4_f32(false, a, false, b,
                                                    (short)0, acc, false, false);
    }
#pragma unroll
    for (int r = 0; r < 8; ++r) {
        const int row = bm + r + 8 * half;
        const int col = on + l16;
        const size_t idx = (size_t)row * O + col;
        const float rn = KAPPA * ro[idx] + acc[r];
        ro[idx] = rn;
        out[((size_t)row * T + t) * O + col] = rn;
    }
}

// ---------------------------------------------------------------------------
// Launch: graph-capture safe (stream-only, no sync/alloc). Deterministic:
// all carried state re-zeroed at entry.
// ---------------------------------------------------------------------------
extern "C" void kernel_launch(void* const* d_in, const int* in_sizes, int n_in,
                              void* d_out, int out_size, void* d_ws, size_t ws_size,
                              hipStream_t stream) {
    (void)in_sizes; (void)n_in; (void)out_size; (void)ws_size;
    const float* x_seq = (const float*)d_in[0];   // [B,T,I]
    const float* W_in  = (const float*)d_in[1];   // [H,I]
    const float* W_rec = (const float*)d_in[2];   // [H,H]
    const float* W_out = (const float*)d_in[3];   // [O,H]

    // d_out = readout_stack [B,T,O] ++ fe_in [B,H,I] ++ fe_rec [B,H,H]
    float* out_ro = (float*)d_out;
    float* fe_in  = out_ro + (size_t)B * T * O;   // +2,097,152
    float* fe_rec = fe_in + (size_t)B * H * I;    // +8,388,608

    // Workspace carve-up (all sizes multiples of 4 floats; ~754 KB total).
    float* ws     = (float*)d_ws;
    float* v      = ws;                 // B*H
    float* psi    = v      + B * H;     // B*H
    float* tr_in  = psi    + B * H;     // B*I
    float* tr_rec = tr_in  + B * I;     // B*H
    float* ro     = tr_rec + B * H;     // B*O
    float* z0     = ro     + B * O;     // B*H
    float* z1     = z0     + B * H;     // B*H
    const int ws_floats = 5 * B * H + B * I + B * O;   // 188,416

    lif_zero<<<64, 256, 0, stream>>>(ws, ws_floats);
    lif_zero<<<2048, 256, 0, stream>>>(fe_in, B * H * I + B * H * H);  // contiguous

    for (int t = 0; t < T; ++t) {
        float* zc = (t & 1) ? z1 : z0;  // z_prev
        float* zn = (t & 1) ? z0 : z1;  // z_next
        // 128 waves: (B/16)*(H/16) tiles
        lif_gemm_neuron<<<16, 256, 0, stream>>>(x_seq, W_in, W_rec, zc, zn, v, psi, t);
        lif_trace<<<192, 256, 0, stream>>>(x_seq, zc, tr_in, tr_rec, t);
        lif_fe<<<6144, 256, 0, stream>>>(fe_in, fe_rec, psi, tr_in, tr_rec);
        // 32 waves: (B/16)*(O/16) tiles
        lif_readout<<<4, 256, 0, stream>>>(zn, W_out, ro, out_ro, t);
    }
}